// kernel_diff_19026705121507
// MI455X (gfx1250) — compile-verified
//
#include <hip/hip_runtime.h>

#define AS1 __attribute__((address_space(1)))
#define AS3 __attribute__((address_space(3)))

typedef int v4i __attribute__((ext_vector_type(4)));

// ---- async global -> LDS path (gfx1250) with guarded fallback ----
#if __has_builtin(__builtin_amdgcn_global_load_async_to_lds_b128)
#define HAVE_ASYNC_LDS 1
#define ASYNC_COPY16(gp, lp) \
    __builtin_amdgcn_global_load_async_to_lds_b128((AS1 v4i*)(gp), (AS3 v4i*)(lp), 0, 0)
#else
#define HAVE_ASYNC_LDS 0
#define ASYNC_COPY16(gp, lp) (*(float4*)(lp) = *(const float4*)(gp))
#endif

static __device__ __forceinline__ void wait_async_zero() {
#if HAVE_ASYNC_LDS
#if __has_builtin(__builtin_amdgcn_s_wait_asynccnt)
    __builtin_amdgcn_s_wait_asynccnt(0);
#else
    asm volatile("s_wait_asynccnt 0" ::: "memory");
#endif
#endif
}

constexpr int N3      = 128;        // D = H = W
constexpr int TD      = 4;          // tile depth  (outputs)
constexpr int TH      = 8;          // tile height (outputs)
constexpr int LD      = TD + 2;     // 6 staged planes
constexpr int LH      = TH + 2;     // 10 staged rows per plane
constexpr int ROWS    = LD * LH;    // 60 rows of 128 floats
constexpr int THREADS = 256;
constexpr int CHUNKS  = ROWS * (N3 / 4);        // 1920 x 16B copies
constexpr int CITERS  = (CHUNKS + THREADS - 1) / THREADS;   // 8

__global__ __launch_bounds__(THREADS)
void stencil3d_wrap_kernel(const float* __restrict__ x,
                           const float* __restrict__ kern,
                           float* __restrict__ out)
{
    __shared__ float tile[ROWS * N3];   // 30720 bytes

    const int tid = threadIdx.x;
    const int blk = blockIdx.x;              // 4096 blocks
    const int n   = blk >> 9;                // / (32*16)
    const int rem = blk & 511;
    const int d0  = (rem >> 4) * TD;
    const int h0  = (rem & 15) * TH;

    const size_t nbase = (size_t)n * N3 * N3 * N3;

    // ---- stage (TD+2)x(TH+2)x128 tile into LDS, circular wrap on d/h ----
    // c = tid + i*256; since 256 % 32 == 0 the w-chunk is iteration-invariant.
    const int cw   = (tid & 31) << 2;        // w start: 0..124 (constant)
    const int row0 = tid >> 5;               // 0..7
#pragma unroll
    for (int i = 0; i < CITERS; ++i) {
        const int row = row0 + i * (THREADS / 32);   // +8 per step
        if (row < ROWS) {                            // only last step masks
            const int dz = row / LH;                 // 0..5
            const int dy = row - dz * LH;            // 0..9
            const int ds = (d0 + dz + N3 - 1) & (N3 - 1);
            const int hs = (h0 + dy + N3 - 1) & (N3 - 1);
            const float* g = x + nbase + ((size_t)ds * N3 + hs) * N3 + cw;
            float*       l = &tile[row * N3 + cw];
            ASYNC_COPY16(g, l);
        }
    }
    wait_async_zero();
    __syncthreads();

    // 27 stencil weights: uniform address -> scalar loads
    float kw[27];
#pragma unroll
    for (int i = 0; i < 27; ++i) kw[i] = kern[i];

    // ---- each thread: 4 iterations x 4 consecutive w outputs ----
#pragma unroll
    for (int it = 0; it < 4; ++it) {
        const int o4 = tid + it * THREADS;       // 0..1023
        const int w4 = (o4 & 31) << 2;           // 0..124
        const int h  = (o4 >> 5) & (TH - 1);     // 0..7
        const int d  = o4 >> 8;                  // 0..3

        float acc0 = 0.f, acc1 = 0.f, acc2 = 0.f, acc3 = 0.f;
#pragma unroll
        for (int kd = 0; kd < 3; ++kd) {
#pragma unroll
            for (int kh = 0; kh < 3; ++kh) {
                const float* row = &tile[((d + kd) * LH + (h + kh)) * N3];
                const float4 m   = *(const float4*)&row[w4];             // ds_load_b128
                const float left  = row[(w4 + N3 - 1) & (N3 - 1)];       // wrap W-
                const float right = row[(w4 + 4)      & (N3 - 1)];       // wrap W+
                const int  kb = (kd * 3 + kh) * 3;
                const float c0 = kw[kb + 0], c1 = kw[kb + 1], c2 = kw[kb + 2];
                acc0 = fmaf(c0, left, fmaf(c1, m.x, fmaf(c2, m.y, acc0)));
                acc1 = fmaf(c0, m.x,  fmaf(c1, m.y, fmaf(c2, m.z, acc1)));
                acc2 = fmaf(c0, m.y,  fmaf(c1, m.z, fmaf(c2, m.w, acc2)));
                acc3 = fmaf(c0, m.z,  fmaf(c1, m.w, fmaf(c2, right, acc3)));
            }
        }
        const size_t off = nbase + ((size_t)(d0 + d) * N3 + (h0 + h)) * N3 + w4;
        *(float4*)&out[off] = make_float4(acc0, acc1, acc2, acc3);   // 16B store
    }
}

extern "C" void kernel_launch(void* const* d_in, const int* in_sizes, int n_in,
                              void* d_out, int out_size, void* d_ws, size_t ws_size,
                              hipStream_t stream) {
    (void)in_sizes; (void)n_in; (void)out_size; (void)d_ws; (void)ws_size;
    const float* x    = (const float*)d_in[0];   // [8,1,128,128,128] fp32
    const float* kern = (const float*)d_in[1];   // [3,3,3] fp32
    float* out        = (float*)d_out;           // [8,1,128,128,128] fp32

    const int nblocks = 8 * (N3 / TD) * (N3 / TH);   // 4096
    stencil3d_wrap_kernel<<<dim3(nblocks), dim3(THREADS), 0, stream>>>(x, kern, out);
}